// FasterRCNN_1846835937542
// MI455X (gfx1250) — compile-verified
//
#include <hip/hip_runtime.h>
#include <hip/hip_bf16.h>
#include <stdint.h>

#define BATCH 2
#define NANCH 8192          // 2^13
#define MASKW (NANCH / 32)  // 256 words per row

typedef float v2f __attribute__((ext_vector_type(2)));
typedef float v8f __attribute__((ext_vector_type(8)));

__device__ __forceinline__ unsigned float_sort_key(float f) {
  unsigned u = __float_as_uint(f);
  return (u & 0x80000000u) ? ~u : (u | 0x80000000u);  // monotonic float->uint
}

// ---------------------------------------------------------------------------
// Kernel 1: sigmoid + SSD decode + sort-key pack (orig order)
// ---------------------------------------------------------------------------
__global__ void decode_kernel(const float* __restrict__ offsets,
                              const float* __restrict__ labels,
                              const float* __restrict__ anchors,
                              unsigned long long* __restrict__ keys,
                              float* __restrict__ boxes,
                              float* __restrict__ probs) {
  int idx = blockIdx.x * blockDim.x + threadIdx.x;
  if (idx >= BATCH * NANCH) return;
  int i = idx & (NANCH - 1);

  float ax1 = anchors[i * 4 + 0], ay1 = anchors[i * 4 + 1];
  float ax2 = anchors[i * 4 + 2], ay2 = anchors[i * 4 + 3];
  float pcx = 0.5f * (ax1 + ax2), pcy = 0.5f * (ay1 + ay2);
  float pw = ax2 - ax1, ph = ay2 - ay1;

  float gcx = offsets[idx * 4 + 0], gcy = offsets[idx * 4 + 1];
  float gw  = offsets[idx * 4 + 2], gh  = offsets[idx * 4 + 3];

  float cx = gcx * pw * 0.1f + pcx;
  float cy = gcy * ph * 0.1f + pcy;
  float w  = expf(gw * 0.2f) * pw;
  float h  = expf(gh * 0.2f) * ph;

  boxes[idx * 4 + 0] = cx - 0.5f * w;
  boxes[idx * 4 + 1] = cy - 0.5f * h;
  boxes[idx * 4 + 2] = cx + 0.5f * w;
  boxes[idx * 4 + 3] = cy + 0.5f * h;

  float logit = labels[idx];
  float p = 1.0f / (1.0f + expf(-logit));
  probs[idx] = p;

  bool valid = p > 0.5f;
  float skey = valid ? p : -1.0f;
  // descending sort on (score, then ascending original index) -> stable argsort
  keys[idx] = ((unsigned long long)float_sort_key(skey) << 32) |
              (unsigned long long)(0xFFFFFFFFu - (unsigned)i);
}

// ---------------------------------------------------------------------------
// Kernel 2: in-LDS bitonic sort of 8192 u64 keys, descending. 64KB LDS/WGP.
// ---------------------------------------------------------------------------
__global__ __launch_bounds__(1024) void sort_kernel(unsigned long long* __restrict__ keys) {
  __shared__ unsigned long long sh[NANCH];
  const int b = blockIdx.x;
  const int t = threadIdx.x;
  unsigned long long* kb = keys + ((size_t)b << 13);

#pragma unroll
  for (int e = 0; e < NANCH / 1024; ++e) sh[t + (e << 10)] = kb[t + (e << 10)];

  for (int k = 2; k <= NANCH; k <<= 1) {
    for (int j = k >> 1; j > 0; j >>= 1) {
      __syncthreads();
#pragma unroll
      for (int e = 0; e < NANCH / 1024; ++e) {
        int idx = t + (e << 10);
        int ixj = idx ^ j;
        if (ixj > idx) {
          bool desc = (idx & k) == 0;  // final merge (k==N) is fully descending
          unsigned long long a = sh[idx], c = sh[ixj];
          bool swp = desc ? (a < c) : (a > c);
          if (swp) { sh[idx] = c; sh[ixj] = a; }
        }
      }
    }
  }
  __syncthreads();
#pragma unroll
  for (int e = 0; e < NANCH / 1024; ++e) kb[t + (e << 10)] = sh[t + (e << 10)];
}

// ---------------------------------------------------------------------------
// Kernel 3: gather into sorted order; precompute areas
// ---------------------------------------------------------------------------
__global__ void gather_kernel(const unsigned long long* __restrict__ keys,
                              const float* __restrict__ boxes,
                              const float* __restrict__ probs,
                              float* __restrict__ sboxes,
                              float* __restrict__ sscores,
                              float* __restrict__ sarea) {
  int idx = blockIdx.x * blockDim.x + threadIdx.x;
  if (idx >= BATCH * NANCH) return;
  int b = idx >> 13;
  unsigned long long k = keys[idx];
  int orig = (int)(0xFFFFFFFFu - (unsigned)(k & 0xFFFFFFFFull));
  int src = (b << 13) + orig;
  float x1 = boxes[src * 4 + 0], y1 = boxes[src * 4 + 1];
  float x2 = boxes[src * 4 + 2], y2 = boxes[src * 4 + 3];
  sboxes[idx * 4 + 0] = x1;
  sboxes[idx * 4 + 1] = y1;
  sboxes[idx * 4 + 2] = x2;
  sboxes[idx * 4 + 3] = y2;
  sscores[idx] = probs[src];
  sarea[idx] = (x2 - x1) * (y2 - y1);
}

// ---------------------------------------------------------------------------
// Kernel 4: overlap bit-matrix, 16-row strip per block. WMMA computes the
// pairwise (area_i + area_j) term of the IoU denominator per 16x16 tile:
//   acc = A_area x (1/4) + (1/4) x B_area^T   (operands replicated over K,
//   so result is invariant to the K-slot packing of the B matrix).
// ---------------------------------------------------------------------------
__global__ __launch_bounds__(256) void mask_kernel(const float* __restrict__ sboxes,
                                                   const float* __restrict__ sarea,
                                                   unsigned* __restrict__ mask) {
  const int b    = blockIdx.y;
  const int rt   = blockIdx.x;   // 16-row tile index, 0..511
  const int t    = threadIdx.x;
  const int lane = t & 31;
  const int wave = t >> 5;
  const int m    = lane & 15;
  const bool hi  = lane >= 16;

  __shared__ float rbx1[16], rby1[16], rbx2[16], rby2[16], rar[16];
  if (t < 16) {
    int row = (b << 13) + (rt << 4) + t;
    rbx1[t] = sboxes[row * 4 + 0];
    rby1[t] = sboxes[row * 4 + 1];
    rbx2[t] = sboxes[row * 4 + 2];
    rby2[t] = sboxes[row * 4 + 3];
    rar[t]  = sarea[row];
  }
  __syncthreads();

  float arM = rar[m];                // A rows: area of row M, all K slots
  v2f a_area = {arM, arM};
  v2f a_q    = {0.25f, 0.25f};

  unsigned* mb = mask + (size_t)b * NANCH * MASKW;

  for (int cb = wave * 32; cb < NANCH; cb += 256) {
    unsigned ball[2][8];
#pragma unroll
    for (int u = 0; u < 2; ++u) {
      int col = cb + u * 16 + m;
      const float* cp = sboxes + (((size_t)(b << 13) + col) << 2);
      float cx1 = cp[0], cy1 = cp[1], cx2 = cp[2], cy2 = cp[3];
      float ac = sarea[(b << 13) + col];
      v2f b_area = {ac, ac};

      v8f acc = {0.f, 0.f, 0.f, 0.f, 0.f, 0.f, 0.f, 0.f};
      // acc[M][N] = sum_k area_row[M] * 0.25 = area_row[M]
      acc = __builtin_amdgcn_wmma_f32_16x16x4_f32(false, a_area, false, a_q,
                                                  (short)0, acc, false, false);
      // acc[M][N] += sum_k 0.25 * area_col[N] = area_row[M] + area_col[N]
      acc = __builtin_amdgcn_wmma_f32_16x16x4_f32(false, a_q, false, b_area,
                                                  (short)0, acc, false, false);

#pragma unroll
      for (int r = 0; r < 8; ++r) {
        int rl = r + (hi ? 8 : 0);  // D layout: VGPR r -> rows r (lanes 0-15), r+8 (16-31)
        float ix = fminf(rbx2[rl], cx2) - fmaxf(rbx1[rl], cx1);
        float iy = fminf(rby2[rl], cy2) - fmaxf(rby1[rl], cy1);
        ix = fmaxf(ix, 0.0f);
        iy = fmaxf(iy, 0.0f);
        float inter = ix * iy;
        float iou = inter / (acc[r] - inter);
        ball[u][r] = (unsigned)__ballot(iou > 0.5f);
      }
    }
    if (lane == 0) {
      int w = cb >> 5;
#pragma unroll
      for (int r = 0; r < 8; ++r) {
        unsigned b0 = ball[0][r], b1 = ball[1][r];
        mb[(size_t)((rt << 4) + r) * MASKW + w]     = (b0 & 0xFFFFu) | (b1 << 16);
        mb[(size_t)((rt << 4) + 8 + r) * MASKW + w] = (b0 >> 16) | (b1 & 0xFFFF0000u);
      }
    }
  }
}

// ---------------------------------------------------------------------------
// Kernel 5: sequential greedy suppression scan (exact reference semantics)
// + output write. One 256-thread block per image; suppress mask in LDS.
// ---------------------------------------------------------------------------
__global__ __launch_bounds__(256) void scan_kernel(const float* __restrict__ sboxes,
                                                   const float* __restrict__ sscores,
                                                   const unsigned* __restrict__ mask,
                                                   float* __restrict__ out) {
  const int b = blockIdx.x;
  const int t = threadIdx.x;
  __shared__ unsigned sup[MASKW];

  const float* sc = sscores + (b << 13);
  unsigned v = 0;
#pragma unroll
  for (int bit = 0; bit < 32; ++bit) {
    if (!(sc[t * 32 + bit] > 0.5f)) v |= (1u << bit);  // suppress0 = ~valid
  }
  sup[t] = v;

  const unsigned* mb = mask + (size_t)b * NANCH * MASKW;
  for (int i = 0; i < NANCH; ++i) {
    __syncthreads();
    if (i + 8 < NANCH)
      __builtin_prefetch((const void*)(mb + (size_t)(i + 8) * MASKW + t), 0, 0);
    unsigned sw = sup[i >> 5];
    bool skip = (sw >> (i & 31)) & 1u;   // uniform across block
    __syncthreads();
    if (!skip) {
      unsigned nv = sup[t] | mb[(size_t)i * MASKW + t];
      if (t == (i >> 5)) nv &= ~(1u << (i & 31));  // keep self
      sup[t] = nv;
    }
  }
  __syncthreads();

  const float* sb = sboxes + (((size_t)b << 13) << 2);
  for (int s = t; s < NANCH; s += 256) {
    bool valid = sc[s] > 0.5f;
    bool kept = valid && !((sup[s >> 5] >> (s & 31)) & 1u);
    size_t ob = ((size_t)(b << 13) + s) << 2;
    out[ob + 0] = kept ? sb[s * 4 + 0] : 0.0f;
    out[ob + 1] = kept ? sb[s * 4 + 1] : 0.0f;
    out[ob + 2] = kept ? sb[s * 4 + 2] : 0.0f;
    out[ob + 3] = kept ? sb[s * 4 + 3] : 0.0f;
    out[(size_t)BATCH * NANCH * 4 + (b << 13) + s] = kept ? sc[s] : 0.0f;
    out[(size_t)BATCH * NANCH * 5 + (b << 13) + s] = kept ? 1.0f : 0.0f;
  }
}

// ---------------------------------------------------------------------------
extern "C" void kernel_launch(void* const* d_in, const int* in_sizes, int n_in,
                              void* d_out, int out_size, void* d_ws, size_t ws_size,
                              hipStream_t stream) {
  (void)in_sizes; (void)n_in; (void)out_size; (void)ws_size;
  const float* offsets = (const float*)d_in[0];  // [B,N,4]
  const float* labels  = (const float*)d_in[1];  // [B,N,1]
  const float* anchors = (const float*)d_in[2];  // [N,4]
  float* out = (float*)d_out;

  char* ws = (char*)d_ws;
  unsigned long long* keys = (unsigned long long*)(ws);        // 128 KB
  float* boxes   = (float*)(ws + (1u << 17));                  // 256 KB
  float* probs   = (float*)(ws + 393216);                      // 64 KB
  float* sboxes  = (float*)(ws + 458752);                      // 256 KB
  float* sscores = (float*)(ws + 720896);                      // 64 KB
  float* sarea   = (float*)(ws + 786432);                      // 64 KB
  unsigned* mask = (unsigned*)(ws + 1048576);                  // 16 MB bitmatrix

  const int total = BATCH * NANCH;
  decode_kernel<<<dim3(total / 256), 256, 0, stream>>>(offsets, labels, anchors,
                                                       keys, boxes, probs);
  sort_kernel<<<dim3(BATCH), 1024, 0, stream>>>(keys);
  gather_kernel<<<dim3(total / 256), 256, 0, stream>>>(keys, boxes, probs,
                                                       sboxes, sscores, sarea);
  mask_kernel<<<dim3(NANCH / 16, BATCH), 256, 0, stream>>>(sboxes, sarea, mask);
  scan_kernel<<<dim3(BATCH), 256, 0, stream>>>(sboxes, sscores, mask, out);
}